// CellSpatialNet_77403900609088
// MI455X (gfx1250) — compile-verified
//
#include <hip/hip_runtime.h>
#include <hip/hip_bf16.h>

#define N_NODES 20480
#define N_EDGES 327680
#define N_BATCH 16
#define CH_OUT  64
#define N_ET    36

typedef __attribute__((ext_vector_type(2))) float v2f;
typedef __attribute__((ext_vector_type(8))) float v8f;

// ---------------------------------------------------------------------------
// Zero-fill scratch regions (deterministic init each call)
// ---------------------------------------------------------------------------
__global__ void fill_zero(float* __restrict__ p, int n) {
    int i = blockIdx.x * blockDim.x + threadIdx.x;
    if (i < n) p[i] = 0.0f;
}

// ---------------------------------------------------------------------------
// Fused EdgeNN + per-edge matvec + scatter-add, LDS-staged tables.
// All parameter tables (emb[ET,D], wh/wg[2,D], bh/bg[D]) are staged into
// CDNA5's 320KB LDS once per persistent block, then a grid-stride loop
// processes (edge, 4-channel-group) work items with float4 ds loads.
//   W[j] = relu(emb[et][j]*(f0*wh0[j]+f1*wh1[j]+bh[j]) + f0*wg0[j]+f1*wg1[j]+bg[j])
//   msg[o] = sum_i h[src][i] * W[i*COUT+o]
// ---------------------------------------------------------------------------
template<int CIN, int COUT>
__global__ void __launch_bounds__(256)
edge_msg_kernel(const int* __restrict__ edge_index,
                const int* __restrict__ edge_type,
                const float* __restrict__ edge_attr,
                const float* __restrict__ emb,
                const float* __restrict__ wh,
                const float* __restrict__ bh,
                const float* __restrict__ wg,
                const float* __restrict__ bg,
                const float* __restrict__ h,
                float* __restrict__ agg,
                float* __restrict__ deg) {
    constexpr int D  = CIN * COUT;
    constexpr int DQ = D / 4;             // float4 quads per table row
    constexpr int OG = COUT / 4;          // 4-channel groups per edge

    __shared__ float4 s_emb[N_ET * DQ];
    __shared__ float4 s_wh[2 * DQ];
    __shared__ float4 s_wg[2 * DQ];
    __shared__ float4 s_bh[DQ];
    __shared__ float4 s_bg[DQ];

    const float4* g_emb = (const float4*)emb;
    const float4* g_wh  = (const float4*)wh;
    const float4* g_wg  = (const float4*)wg;
    const float4* g_bh  = (const float4*)bh;
    const float4* g_bg  = (const float4*)bg;
    for (int i = threadIdx.x; i < N_ET * DQ; i += blockDim.x) s_emb[i] = g_emb[i];
    for (int i = threadIdx.x; i < 2 * DQ; i += blockDim.x) {
        s_wh[i] = g_wh[i];
        s_wg[i] = g_wg[i];
    }
    for (int i = threadIdx.x; i < DQ; i += blockDim.x) {
        s_bh[i] = g_bh[i];
        s_bg[i] = g_bg[i];
    }
    __syncthreads();

    const int total  = N_EDGES * OG;
    const int stride = gridDim.x * blockDim.x;
    for (int idx = blockIdx.x * blockDim.x + threadIdx.x; idx < total; idx += stride) {
        int e  = idx / OG;
        int og = idx - e * OG;

        int src = edge_index[e];
        int dst = edge_index[N_EDGES + e];
        int et  = edge_type[e];
        float f0 = edge_attr[2 * e + 0];
        float f1 = edge_attr[2 * e + 1];
        const float* hr = h + (size_t)src * CIN;
        const float4* er = s_emb + et * DQ;

        float4 msg = make_float4(0.f, 0.f, 0.f, 0.f);
        #pragma unroll
        for (int i = 0; i < CIN; ++i) {
            int jq = i * (COUT / 4) + og;
            float4 em = er[jq];
            float4 w0 = s_wh[jq], w1 = s_wh[DQ + jq];
            float4 g0 = s_wg[jq], g1 = s_wg[DQ + jq];
            float4 b0 = s_bh[jq], b1 = s_bg[jq];
            float hi = hr[i];
            float ax = fmaf(f0, w0.x, fmaf(f1, w1.x, b0.x));
            float ay = fmaf(f0, w0.y, fmaf(f1, w1.y, b0.y));
            float az = fmaf(f0, w0.z, fmaf(f1, w1.z, b0.z));
            float aw = fmaf(f0, w0.w, fmaf(f1, w1.w, b0.w));
            float gx = fmaf(f0, g0.x, fmaf(f1, g1.x, b1.x));
            float gy = fmaf(f0, g0.y, fmaf(f1, g1.y, b1.y));
            float gz = fmaf(f0, g0.z, fmaf(f1, g1.z, b1.z));
            float gw = fmaf(f0, g0.w, fmaf(f1, g1.w, b1.w));
            msg.x = fmaf(hi, fmaxf(fmaf(em.x, ax, gx), 0.f), msg.x);
            msg.y = fmaf(hi, fmaxf(fmaf(em.y, ay, gy), 0.f), msg.y);
            msg.z = fmaf(hi, fmaxf(fmaf(em.z, az, gz), 0.f), msg.z);
            msg.w = fmaf(hi, fmaxf(fmaf(em.w, aw, gw), 0.f), msg.w);
        }
        float* ap = agg + (size_t)dst * COUT + og * 4;
        atomicAdd(ap + 0, msg.x);
        atomicAdd(ap + 1, msg.y);
        atomicAdd(ap + 2, msg.z);
        atomicAdd(ap + 3, msg.w);
        if (og == 0) atomicAdd(&deg[dst], 1.0f);
    }
}

// ---------------------------------------------------------------------------
// Node update: out = relu(agg/max(deg,1) + h @ root + bias)
// Root GEMM via V_WMMA_F32_16X16X4_F32, one wave per 16x16 output tile.
// f32 A 16x4 layout:  lane<16 -> M=lane,   VGPRs K0,K1 ; lane>=16 -> K2,K3
// f32 B 4x16 mirrors A with N striped across lanes.
// C/D: VGPR j holds (M = j + 8*(lane>=16), N = lane&15).
// ---------------------------------------------------------------------------
template<int CIN, int COUT>
__global__ void __launch_bounds__(32)
node_root_wmma_kernel(const float* __restrict__ h,
                      const float* __restrict__ root,
                      const float* __restrict__ bias,
                      const float* __restrict__ agg,
                      const float* __restrict__ deg,
                      float* __restrict__ out) {
    constexpr int NT = (COUT + 15) / 16;
    int tile = blockIdx.x;
    int mt = tile / NT;
    int nt = tile - mt * NT;
    int lane = threadIdx.x;          // one full wave, EXEC all-ones
    int half = lane >> 4;
    int lm   = lane & 15;
    int node = mt * 16 + lm;
    int ncol = nt * 16 + lm;
    int ccl  = (ncol < COUT) ? ncol : 0;          // branchless in-bounds clamp
    float bmask = (ncol < COUT) ? 1.0f : 0.0f;

    v8f c = {};
    #pragma unroll
    for (int k0 = 0; k0 < CIN; k0 += 4) {
        int ka = k0 + half * 2;
        v2f a, b;
        a.x = h[(size_t)node * CIN + ka];
        a.y = h[(size_t)node * CIN + ka + 1];
        b.x = root[ka * COUT + ccl] * bmask;
        b.y = root[(ka + 1) * COUT + ccl] * bmask;
        c = __builtin_amdgcn_wmma_f32_16x16x4_f32(
                false, a, false, b, (short)0, c, false, false);
    }

    #pragma unroll
    for (int j = 0; j < 8; ++j) {
        int row = mt * 16 + j + half * 8;
        int col = nt * 16 + lm;
        if (col < COUT) {
            float dn = fmaxf(deg[row], 1.0f);
            float v = agg[(size_t)row * COUT + col] / dn + c[j] + bias[col];
            out[(size_t)row * COUT + col] = fmaxf(v, 0.0f);
        }
    }
}

// ---------------------------------------------------------------------------
// Gated pooling: sum h over nodes with cell_type==1 per graph, plus counts.
// ---------------------------------------------------------------------------
__global__ void pool_kernel(const float* __restrict__ h,
                            const int* __restrict__ cell_type,
                            const int* __restrict__ batch,
                            float* __restrict__ pooled,
                            float* __restrict__ cnt,
                            int n, int co) {
    int idx = blockIdx.x * blockDim.x + threadIdx.x;
    if (idx >= n * co) return;
    int node = idx / co;
    int c = idx - node * co;
    if (cell_type[node] == 1) {
        int b = batch[node];
        atomicAdd(&pooled[b * co + c], h[(size_t)node * co + c]);
        if (c == 0) atomicAdd(&cnt[b], 1.0f);
    }
}

// ---------------------------------------------------------------------------
// Final classifier: sigmoid(mean_pooled @ clf_w + clf_b), B outputs.
// ---------------------------------------------------------------------------
__global__ void clf_kernel(const float* __restrict__ pooled,
                           const float* __restrict__ cnt,
                           const float* __restrict__ clf_w,
                           const float* __restrict__ clf_b,
                           float* __restrict__ out,
                           int nb, int co) {
    int b = blockIdx.x * blockDim.x + threadIdx.x;
    if (b >= nb) return;
    float dn = fmaxf(cnt[b], 1.0f);
    float acc = clf_b[0];
    for (int c = 0; c < co; ++c)
        acc = fmaf(pooled[b * co + c] / dn, clf_w[c], acc);
    out[b] = 1.0f / (1.0f + expf(-acc));
}

// ---------------------------------------------------------------------------
extern "C" void kernel_launch(void* const* d_in, const int* in_sizes, int n_in,
                              void* d_out, int out_size, void* d_ws, size_t ws_size,
                              hipStream_t stream) {
    const float* x          = (const float*)d_in[0];
    const int*   edge_index = (const int*)d_in[1];
    const int*   edge_type  = (const int*)d_in[2];
    const float* edge_attr  = (const float*)d_in[3];
    const int*   cell_type  = (const int*)d_in[4];
    const int*   batch      = (const int*)d_in[5];
    const float* clf_w      = (const float*)d_in[34];
    const float* clf_b      = (const float*)d_in[35];
    float* out = (float*)d_out;

    // Workspace carve-up (floats)
    float* ws = (float*)d_ws;
    float* hA     = ws;                                   // N*64
    float* hB     = hA + (size_t)N_NODES * 64;            // N*64
    float* agg    = hB + (size_t)N_NODES * 64;            // N*64
    float* deg    = agg + (size_t)N_NODES * 64;           // N
    float* pooled = deg + N_NODES;                        // B*CO
    float* cnt    = pooled + N_BATCH * CH_OUT;            // B

    const int couts[4] = {8, 8, 8, 64};
    const float* P[4][7];
    for (int l = 0; l < 4; ++l)
        for (int k = 0; k < 7; ++k)
            P[l][k] = (const float*)d_in[6 + 7 * l + k];   // emb,wh,bh,wg,bg,root,bias

    const int EDGE_BLOCKS = 1024;
    const int M_TILES = N_NODES / 16;

    const float* h_in = x;
    float* bufs[2] = {hA, hB};

    for (int l = 0; l < 4; ++l) {
        int cout = couts[l];
        float* h_out = bufs[l & 1];

        int naz = N_NODES * cout;
        fill_zero<<<(naz + 255) / 256, 256, 0, stream>>>(agg, naz);
        fill_zero<<<(N_NODES + 255) / 256, 256, 0, stream>>>(deg, N_NODES);

        switch (l) {
        case 0:
            edge_msg_kernel<16, 8><<<EDGE_BLOCKS, 256, 0, stream>>>(
                edge_index, edge_type, edge_attr,
                P[0][0], P[0][1], P[0][2], P[0][3], P[0][4], h_in, agg, deg);
            node_root_wmma_kernel<16, 8><<<M_TILES, 32, 0, stream>>>(
                h_in, P[0][5], P[0][6], agg, deg, h_out);
            break;
        case 1:
            edge_msg_kernel<8, 8><<<EDGE_BLOCKS, 256, 0, stream>>>(
                edge_index, edge_type, edge_attr,
                P[1][0], P[1][1], P[1][2], P[1][3], P[1][4], h_in, agg, deg);
            node_root_wmma_kernel<8, 8><<<M_TILES, 32, 0, stream>>>(
                h_in, P[1][5], P[1][6], agg, deg, h_out);
            break;
        case 2:
            edge_msg_kernel<8, 8><<<EDGE_BLOCKS, 256, 0, stream>>>(
                edge_index, edge_type, edge_attr,
                P[2][0], P[2][1], P[2][2], P[2][3], P[2][4], h_in, agg, deg);
            node_root_wmma_kernel<8, 8><<<M_TILES, 32, 0, stream>>>(
                h_in, P[2][5], P[2][6], agg, deg, h_out);
            break;
        case 3:
            edge_msg_kernel<8, 64><<<EDGE_BLOCKS, 256, 0, stream>>>(
                edge_index, edge_type, edge_attr,
                P[3][0], P[3][1], P[3][2], P[3][3], P[3][4], h_in, agg, deg);
            node_root_wmma_kernel<8, 64><<<M_TILES * 4, 32, 0, stream>>>(
                h_in, P[3][5], P[3][6], agg, deg, h_out);
            break;
        }
        h_in = h_out;
    }

    // pooling (h_in == hB after layer 4, stride CH_OUT)
    int npz = N_BATCH * CH_OUT + N_BATCH;   // pooled + cnt contiguous
    fill_zero<<<(npz + 255) / 256, 256, 0, stream>>>(pooled, npz);
    int npool = N_NODES * CH_OUT;
    pool_kernel<<<(npool + 255) / 256, 256, 0, stream>>>(
        h_in, cell_type, batch, pooled, cnt, N_NODES, CH_OUT);

    clf_kernel<<<1, 32, 0, stream>>>(pooled, cnt, clf_w, clf_b, out,
                                     N_BATCH, CH_OUT);
}